// Lcross_11106785427996
// MI455X (gfx1250) — compile-verified
//
#include <hip/hip_runtime.h>
#include <hip/hip_bf16.h>
#include <stdint.h>

// ---------------------------------------------------------------------------
// Weighted masked NLL loss reduction, MI455X (gfx1250, wave32).
// Memory-bound: ~100.7 MB read once -> ~4.3 us at 23.3 TB/s.
// Streams realinput + labels through LDS with double-buffered
// GLOBAL_LOAD_ASYNC_TO_LDS_B128 (ASYNCcnt-tracked), wave32 shuffle reduce,
// deterministic two-pass global reduction (no float atomics).
// ---------------------------------------------------------------------------

#define THREADS   256
#define VPT       4u          // voxels per thread per tile
#define TILE_VOX  1024u       // voxels per tile (THREADS * VPT)
#define DHW       2097152u    // 128*128*128
#define NCLS      5
#define NBLOCKS   1024
#define TILES_PB  4           // tiles per block; NBLOCKS*TILES_PB*TILE_VOX = 4,194,304
#define PLANE_B   4096u       // bytes per class plane per tile (1024 voxels * 4B)
#define TILE_B    24576u      // 6 planes (5 classes + labels) * 4096 B
#define TILE_F    6144u       // TILE_B in floats

// One wave-wide async copy: each lane moves 16B global -> LDS. ASYNCcnt++.
__device__ __forceinline__ void async_b128(uint32_t lds_byte, uint64_t sbase,
                                           uint32_t voff_bytes) {
  // GVS mode: mem_addr = SGPR64 + VGPR32 offset; vdst VGPR = LDS byte address.
  asm volatile("global_load_async_to_lds_b128 %0, %1, %2"
               :
               : "v"(lds_byte), "v"(voff_bytes), "s"(sbase)
               : "memory");
}

// Stage one tile (5 class planes + labels) into LDS buffer `buf`.
// Each thread stages exactly the bytes it will later consume -> no barrier.
__device__ __forceinline__ void issue_tile(const float* input, const int* label,
                                           uint32_t ldsBase, uint32_t buf,
                                           uint32_t tile, uint32_t t) {
  const uint32_t v0   = tile * TILE_VOX;        // first voxel of tile
  const uint32_t b    = v0 / DHW;               // batch index (tile never splits b)
  const uint32_t s0   = v0 - b * DHW;           // spatial offset within batch
  const uint32_t ldsT = ldsBase + buf * TILE_B + t * 16u;
  const uint32_t e0   = s0 + t * VPT;           // this thread's first element
#pragma unroll
  for (uint32_t c = 0; c < NCLS; ++c) {
    const uint32_t goff = ((b * NCLS + c) * DHW + e0) * 4u;   // bytes (< 2^27)
    async_b128(ldsT + c * PLANE_B, (uint64_t)(uintptr_t)input, goff);
  }
  async_b128(ldsT + NCLS * PLANE_B, (uint64_t)(uintptr_t)label,
             (v0 + t * VPT) * 4u);
}

__global__ __launch_bounds__(THREADS) void lcross_main(
    const float* __restrict__ input, const int* __restrict__ label,
    const float* __restrict__ Wl, float* __restrict__ partials,
    uint32_t* __restrict__ masks) {
  __shared__ __align__(16) float smem[2 * TILE_F];   // 48 KB double buffer
  __shared__ float    wsum[THREADS / 32];
  __shared__ uint32_t wmsk[THREADS / 32];

  const uint32_t t = threadIdx.x;
  // Low 32 bits of the flat pointer to an LDS object == LDS byte offset.
  const uint32_t ldsBase = (uint32_t)(uintptr_t)(void*)&smem[0];

  const float w1 = Wl[0], w2 = Wl[1], w3 = Wl[2], w4 = Wl[3];

  const uint32_t tile0 = (uint32_t)blockIdx.x * TILES_PB;

  // Prologue: stage tile 0 into buffer 0 (6 async ops in flight).
  issue_tile(input, label, ldsBase, 0u, tile0, t);

  float acc = 0.0f;
  uint32_t mask = 0u;

  for (int i = 0; i < TILES_PB; ++i) {
    if (i + 1 < TILES_PB) {
      // WAR guard: our earlier ds reads of this buffer must land before the
      // async engine may overwrite it.
      asm volatile("s_wait_dscnt 0x0" ::: "memory");
      issue_tile(input, label, ldsBase, (uint32_t)((i + 1) & 1),
                 tile0 + (uint32_t)(i + 1), t);
      // 12 async ops outstanding at most; <=6 left means tile i is complete.
      asm volatile("s_wait_asynccnt 0x6" ::: "memory");
    } else {
      asm volatile("s_wait_asynccnt 0x0" ::: "memory");
    }

    const float* base = smem + (uint32_t)(i & 1) * TILE_F;
    const int4 l4 = reinterpret_cast<const int4*>(base + NCLS * 1024u)[t];
    const int ls[4] = {l4.x, l4.y, l4.z, l4.w};
#pragma unroll
    for (uint32_t j = 0; j < VPT; ++j) {
      const int l = ls[j];
      if (l > 0) {
        const float p = base[(uint32_t)l * 1024u + t * VPT + j];
        const float w = (l == 1) ? w1 : (l == 2) ? w2 : (l == 3) ? w3 : w4;
        acc += w * (-__logf(p));          // -ln p  (p >= 1e-6, normal range)
        mask |= (1u << (uint32_t)l);
      }
    }
  }

  // wave32 butterfly reduction (deterministic, fixed order)
#pragma unroll
  for (int off = 16; off > 0; off >>= 1) {
    acc += __shfl_xor(acc, off, 32);
    mask |= (uint32_t)__shfl_xor((int)mask, off, 32);
  }
  if ((t & 31u) == 0u) {
    wsum[t >> 5] = acc;
    wmsk[t >> 5] = mask;
  }
  __syncthreads();
  if (t == 0) {
    float s = 0.0f;
    uint32_t m = 0u;
#pragma unroll
    for (int i = 0; i < THREADS / 32; ++i) { s += wsum[i]; m |= wmsk[i]; }
    partials[blockIdx.x] = s;
    masks[blockIdx.x]    = m;
  }
}

__global__ __launch_bounds__(256) void lcross_final(
    const float* __restrict__ partials, const uint32_t* __restrict__ masks,
    const float* __restrict__ Wl, const float* __restrict__ label_sum,
    float* __restrict__ out) {
  __shared__ double   ts[256];
  __shared__ uint32_t tm[256];
  const int t = threadIdx.x;
  double a = 0.0;
  uint32_t m = 0u;
  for (int i = t; i < NBLOCKS; i += 256) {
    a += (double)partials[i];
    m |= masks[i];
  }
  ts[t] = a;
  tm[t] = m;
  __syncthreads();
  if (t == 0) {
    double s = 0.0;
    uint32_t mm = 0u;
    for (int i = 0; i < 256; ++i) { s += ts[i]; mm |= tm[i]; }
    double plus = 0.0;
#pragma unroll
    for (int c = 1; c < NCLS; ++c)
      if (mm & (1u << c)) plus += (double)Wl[c - 1];
    const double denom = (double)label_sum[0] + (double)label_sum[1] +
                         (double)label_sum[2] + (double)label_sum[3];
    out[0] = (float)((s + plus) / denom);
  }
}

extern "C" void kernel_launch(void* const* d_in, const int* in_sizes, int n_in,
                              void* d_out, int out_size, void* d_ws,
                              size_t ws_size, hipStream_t stream) {
  const float* input     = (const float*)d_in[0];  // [2,5,128,128,128] f32
  const int*   label     = (const int*)d_in[1];    // [2,128,128,128] i32
  const float* Wl        = (const float*)d_in[2];  // [4] f32
  const float* label_sum = (const float*)d_in[3];  // [4] f32
  float*       out       = (float*)d_out;          // scalar f32

  float*    partials = (float*)d_ws;                                   // 1024 f32
  uint32_t* masks    = (uint32_t*)((char*)d_ws + NBLOCKS * sizeof(float)); // 1024 u32

  lcross_main<<<NBLOCKS, THREADS, 0, stream>>>(input, label, Wl, partials, masks);
  lcross_final<<<1, 256, 0, stream>>>(partials, masks, Wl, label_sum, out);
}